// WrapLayer_42880953483790
// MI455X (gfx1250) — compile-verified
//
#include <hip/hip_runtime.h>

// Backward bilinear warp, CDNA5 / gfx1250.
// Layout facts (derived from the reference's reshapes):
//   image  (B,M,N,2) flat == 2B planes of M*N floats
//   flow   (B,M,N,2): flow[b] drives BOTH planes 2b and 2b+1
//   out    (B,M,N,2) flat == 2B planes of M*N floats
// One thread handles (i,j) for a strided set of batches b. Per iteration:
// 1x NT B64 flow load, 8 gather loads (4 offsets x 2 contiguous planes),
// 2 NT B32 stores, plus a global_prefetch of the next batch's flow.
// Grid-stride is an exact multiple of M*N, so (i,j) are loop-invariant and
// the loop carries only pointer bumps (no div/mod anywhere in the loop).

typedef __attribute__((ext_vector_type(2))) float v2f;

#define BDIM   256
#define PASSES 8          // batches per thread step; grid covers b0 in [0, PASSES)

__global__ __launch_bounds__(BDIM) void warp_bilinear_kernel(
    const float* __restrict__ image,
    const float* __restrict__ flow,
    float* __restrict__ out,
    int B)
{
    constexpr int M  = 384;
    constexpr int N  = 384;
    constexpr int MN = M * N;

    // One-time decode (constant divisors -> mulhi+shift, no runtime rcp).
    int t  = blockIdx.x * BDIM + threadIdx.x;   // [0, min(B,PASSES)*MN)
    int j  = t % N;
    int r  = t / N;
    int i  = r % M;
    int b0 = r / M;                             // [0, PASSES)

    const float fj = (float)j;
    const float fi = (float)i;
    const int   pix = i * N + j;                // within-plane offset

    const v2f*   __restrict__ fp = (const v2f*)flow + (size_t)b0 * MN + pix;
    const float* __restrict__ ip = image + (size_t)(2 * b0) * MN;  // plane 2b
    float*       __restrict__ op = out   + (size_t)(2 * b0) * MN + pix;

    const size_t fstep = (size_t)PASSES * MN;       // v2f elements per step
    const size_t istep = (size_t)(2 * PASSES) * MN; // floats per step

    for (int b = b0; b < B; b += PASSES) {
        // Prefetch next batch's flow (global_prefetch_b8, speculative RT).
        if (b + PASSES < B) __builtin_prefetch(fp + fstep, 0, 1);

        // Flow is read exactly once -> non-temporal, keep L2 for the image.
        v2f f = __builtin_nontemporal_load(fp);

        float xs = fj + f.x;
        float ys = fi + f.y;
        float x0f = floorf(xs);
        float y0f = floorf(ys);
        float wx = xs - x0f;
        float wy = ys - y0f;
        int x0 = (int)x0f;
        int y0 = (int)y0f;
        int x1 = x0 + 1;
        int y1 = y0 + 1;

        float omwx = 1.0f - wx, omwy = 1.0f - wy;
        float w00 = omwy * omwx;
        float w01 = omwy * wx;
        float w10 = wy * omwx;
        float w11 = wy * wx;

        // In-range test as a single unsigned compare per coordinate.
        bool vx0 = (unsigned)x0 < (unsigned)N;
        bool vx1 = (unsigned)x1 < (unsigned)N;
        bool vy0 = (unsigned)y0 < (unsigned)M;
        bool vy1 = (unsigned)y1 < (unsigned)M;
        // Zero the weight of invalid corners; gather from clamped (always
        // in-bounds, finite) addresses. Exactly matches where(valid, v*w, 0).
        w00 = (vy0 && vx0) ? w00 : 0.0f;
        w01 = (vy0 && vx1) ? w01 : 0.0f;
        w10 = (vy1 && vx0) ? w10 : 0.0f;
        w11 = (vy1 && vx1) ? w11 : 0.0f;

        int cx0 = min(max(x0, 0), N - 1);   // v_med3_i32
        int cx1 = min(max(x1, 0), N - 1);
        int cy0 = min(max(y0, 0), M - 1);
        int cy1 = min(max(y1, 0), M - 1);

        int rowA = cy0 * N;
        int rowB = cy1 * N;
        int o00 = rowA + cx0;
        int o01 = rowA + cx1;
        int o10 = rowB + cx0;
        int o11 = rowB + cx1;

        // 8 independent gathers; plane-1 reuses plane-0 address VGPRs via the
        // +MN*4 byte immediate offset. (RT: 75MB image resident in 192MB L2.)
        float a00 = ip[o00],      a01 = ip[o01],      a10 = ip[o10],      a11 = ip[o11];
        float b00 = ip[MN + o00], b01 = ip[MN + o01], b10 = ip[MN + o10], b11 = ip[MN + o11];

        // Same association order as the reference, fused.
        float r0 = fmaf(a11, w11, fmaf(a10, w10, fmaf(a01, w01, a00 * w00)));
        float r1 = fmaf(b11, w11, fmaf(b10, w10, fmaf(b01, w01, b00 * w00)));

        __builtin_nontemporal_store(r0, op);       // plane 2b
        __builtin_nontemporal_store(r1, op + MN);  // plane 2b+1

        fp += fstep;
        ip += istep;
        op += istep;
    }
}

extern "C" void kernel_launch(void* const* d_in, const int* in_sizes, int n_in,
                              void* d_out, int out_size, void* d_ws, size_t ws_size,
                              hipStream_t stream) {
    const float* image = (const float*)d_in[0];
    const float* flow  = (const float*)d_in[1];
    float* out = (float*)d_out;

    constexpr int M = 384, N = 384, MN = M * N;
    const int B = in_sizes[0] / (MN * 2);            // 64 for the reference setup

    const int bcov = (B < PASSES) ? B : PASSES;      // b0 coverage
    const int threads = bcov * MN;                   // MN % BDIM == 0 (147456/256)
    const int grid = (threads + BDIM - 1) / BDIM;    // 4608 for B>=8

    hipLaunchKernelGGL(warp_bilinear_kernel, dim3(grid), dim3(BDIM), 0, stream,
                       image, flow, out, B);
}